// DirMax_72816875536667
// MI455X (gfx1250) — compile-verified
//
#include <hip/hip_runtime.h>
#include <stdint.h>

// x (32, 64, 128, 128) f32 -> out (32, 128, 128, 128)
// out channels: [0:64]=x, [64:80]=down, [80:96]=up, [96:112]=right, [112:128]=left

constexpr int N    = 32;
constexpr int C    = 64;
constexpr int H    = 128;
constexpr int W    = 128;
constexpr int PICK = 16;
constexpr int OUTC = C + 4 * PICK;     // 128
constexpr int HW   = H * W;            // 16384 floats = 64 KB
constexpr int HW4  = HW / 4;           // 4096 float4

#define NEG_INF (-__builtin_inff())

__device__ __forceinline__ float4 max4(float4 a, float4 b) {
    return make_float4(fmaxf(a.x, b.x), fmaxf(a.y, b.y),
                       fmaxf(a.z, b.z), fmaxf(a.w, b.w));
}

// One block per (n, c<16) plane. 256 threads = 8 wave32.
// HBM plane is read exactly once via async global->LDS DMA; all five outputs
// (copy + 4 scans) are produced from the LDS-staged tile.
__global__ __launch_bounds__(256)
void dirmax_scan_kernel(const float* __restrict__ x, float* __restrict__ out) {
    __shared__ float4 tile[HW4];                 // 64 KB of the 320 KB WGP LDS

    const int tid   = threadIdx.x;
    const int plane = blockIdx.x;                // 0 .. N*PICK-1
    const int n     = plane >> 4;                // / PICK
    const int c     = plane & (PICK - 1);

    const float* __restrict__ src =
        x + (size_t)(n * C + c) * HW;            // plane base (SADDR)
    float4* __restrict__ cpy4   = (float4*)(out + (size_t)(n * OUTC + c) * HW);
    float4* __restrict__ down4  = (float4*)(out + (size_t)(n * OUTC + C + c) * HW);
    float4* __restrict__ up4    = (float4*)(out + (size_t)(n * OUTC + C + PICK + c) * HW);
    float4* __restrict__ right4 = (float4*)(out + (size_t)(n * OUTC + C + 2 * PICK + c) * HW);
    float4* __restrict__ left4  = (float4*)(out + (size_t)(n * OUTC + C + 3 * PICK + c) * HW);

    // ---- Phase 1: async global -> LDS (ASYNCcnt), 16 x b128 per thread ----
    // GVS mode: LDS address from the vdst VGPR, global address = saddr + vaddr VGPR  (CDNA5 ISA)
    // LDS byte address = low 32 bits of the generic pointer (LDS aperture).
    const unsigned ldsBase = (unsigned)(uintptr_t)(void*)tile;
#pragma unroll
    for (int i = 0; i < HW4 / 256; ++i) {        // 16 async b128 per thread
        const unsigned off = (unsigned)(tid * 16 + i * 4096);   // bytes in plane
        const unsigned lds = ldsBase + off;                     // same layout in LDS
        asm volatile("global_load_async_to_lds_b128 %0, %1, %2"
                     :: "v"(lds), "v"(off), "s"(src)
                     : "memory");
    }
    asm volatile("s_wait_asynccnt 0x0" ::: "memory");
    __syncthreads();

    // ---- Phase 2: pass-through copy served from LDS (no second HBM read) ----
#pragma unroll
    for (int i = 0; i < HW4 / 256; ++i) {
        const int idx = tid + i * 256;
        cpy4[idx] = tile[idx];
    }

    const int wv   = tid >> 5;
    const int lane = tid & 31;
    const float4 ninf4 = make_float4(NEG_INF, NEG_INF, NEG_INF, NEG_INF);

    if (wv == 0) {
        // ---- down: inclusive cummax over h. Lane owns cols 4*lane..4*lane+3.
        float4 run = ninf4;
        for (int h = 0; h < H; ++h) {
            run = max4(run, tile[h * 32 + lane]);
            down4[h * 32 + lane] = run;          // coalesced 512B wave store
        }
    } else if (wv == 1) {
        // ---- up: reverse cummax over h.
        float4 run = ninf4;
        for (int h = H - 1; h >= 0; --h) {
            run = max4(run, tile[h * 32 + lane]);
            up4[h * 32 + lane] = run;
        }
    } else {
        // ---- right/left: one wave per row; wave32 shuffle max-scan.
        for (int r = wv - 2; r < H; r += 6) {
            const float4 v = tile[r * 32 + lane];

            // forward (right): local prefix max over the 4 owned elements
            float4 F;
            F.x = v.x;
            F.y = fmaxf(F.x, v.y);
            F.z = fmaxf(F.y, v.z);
            F.w = fmaxf(F.z, v.w);
            float tot = F.w;
#pragma unroll
            for (int d = 1; d < 32; d <<= 1) {
                const float o = __shfl_up(tot, (unsigned)d, 32);
                if (lane >= d) tot = fmaxf(tot, o);
            }
            float ex = __shfl_up(tot, 1u, 32);
            if (lane == 0) ex = NEG_INF;         // exclusive prefix across lanes
            F.x = fmaxf(F.x, ex);
            F.y = fmaxf(F.y, ex);
            F.z = fmaxf(F.z, ex);
            F.w = fmaxf(F.w, ex);
            right4[r * 32 + lane] = F;

            // backward (left): local suffix max + reverse wave scan
            float4 B;
            B.w = v.w;
            B.z = fmaxf(B.w, v.z);
            B.y = fmaxf(B.z, v.y);
            B.x = fmaxf(B.y, v.x);
            float tb = B.x;
#pragma unroll
            for (int d = 1; d < 32; d <<= 1) {
                const float o = __shfl_down(tb, (unsigned)d, 32);
                if (lane < 32 - d) tb = fmaxf(tb, o);
            }
            float ex2 = __shfl_down(tb, 1u, 32);
            if (lane == 31) ex2 = NEG_INF;       // exclusive suffix across lanes
            B.x = fmaxf(B.x, ex2);
            B.y = fmaxf(B.y, ex2);
            B.z = fmaxf(B.z, ex2);
            B.w = fmaxf(B.w, ex2);
            left4[r * 32 + lane] = B;
        }
    }
}

// Pass-through copy for channels PICK..C-1. 4 float4 per thread: loads are
// clustered in one clause (4 outstanding b128), then 4 coalesced stores.
__global__ __launch_bounds__(256)
void dirmax_copy_kernel(const float* __restrict__ x, float* __restrict__ out) {
    const int base = blockIdx.x * 1024 + threadIdx.x;   // float4 index chunk
    const int per  = (C - PICK) * HW4;                  // 196608 float4 per n

    const float4* __restrict__ src4 = (const float4*)x;
    float4* __restrict__ dst4       = (float4*)out;

    float4 v[4];
    size_t d[4];
#pragma unroll
    for (int k = 0; k < 4; ++k) {
        const int i   = base + k * 256;
        const int n   = i / per;
        const int rem = i - n * per;
        v[k] = src4[(size_t)n * (C * HW4) + (size_t)PICK * HW4 + rem];
        d[k] = (size_t)n * (OUTC * HW4) + (size_t)PICK * HW4 + rem;
    }
#pragma unroll
    for (int k = 0; k < 4; ++k) dst4[d[k]] = v[k];
}

extern "C" void kernel_launch(void* const* d_in, const int* in_sizes, int n_in,
                              void* d_out, int out_size, void* d_ws, size_t ws_size,
                              hipStream_t stream) {
    (void)in_sizes; (void)n_in; (void)out_size; (void)d_ws; (void)ws_size;

    const float* x = (const float*)d_in[0];
    float* out     = (float*)d_out;

    // Kernel A: one block per picked plane (32 * 16 = 512 blocks).
    dirmax_scan_kernel<<<N * PICK, 256, 0, stream>>>(x, out);

    // Kernel B: copy remaining 48 channels per batch; 4 float4 per thread.
    const int copy_f4 = N * (C - PICK) * HW4;           // 6,291,456 float4
    dirmax_copy_kernel<<<copy_f4 / 1024, 256, 0, stream>>>(x, out);
}